// GCN_89928025244111
// MI455X (gfx1250) — compile-verified
//
#include <hip/hip_runtime.h>

// ---------------------------------------------------------------------------
// GCN (3-layer) for MI455X / gfx1250.
// GEMMs use V_WMMA_F32_16X16X4_F32 (full fp32, matches reference precision;
// GEMM cost ~1.4 GFLOP is negligible vs the edge scatter which dominates).
// Aggregations: per-(edge,feature) gather + float atomic scatter; the 13 MB
// node-feature buffers live in the 192 MB L2, only edge streams hit HBM.
// ---------------------------------------------------------------------------

typedef __attribute__((ext_vector_type(2))) float v2f;
typedef __attribute__((ext_vector_type(8))) float v8f;

#define TPB 256

// deg[i] = 1.0 (self-loop weight, fill_value=1.0)
__global__ void k_init_deg(float* __restrict__ deg, int N) {
    int i = blockIdx.x * blockDim.x + threadIdx.x;
    if (i < N) deg[i] = 1.0f;
}

// deg[dst[e]] += w[e]
__global__ void k_edge_deg(const int* __restrict__ dst,
                           const float* __restrict__ w,
                           float* __restrict__ deg, int E) {
    int e = blockIdx.x * blockDim.x + threadIdx.x;
    if (e < E) atomicAdd(&deg[dst[e]], w[e]);
}

// dinv = rsqrt(deg) (deg>0 guaranteed but keep the guard); snorm = dinv^2
__global__ void k_dinv(const float* __restrict__ deg,
                       float* __restrict__ dinv,
                       float* __restrict__ snorm, int N) {
    int i = blockIdx.x * blockDim.x + threadIdx.x;
    if (i < N) {
        float d = deg[i];
        float r = (d > 0.0f) ? __frsqrt_rn(d) : 0.0f;
        dinv[i]  = r;
        snorm[i] = r * r;
    }
}

// norm[e] = dinv[src] * w * dinv[dst]
__global__ void k_norm(const int* __restrict__ src, const int* __restrict__ dst,
                       const float* __restrict__ w, const float* __restrict__ dinv,
                       float* __restrict__ norm, int E) {
    int e = blockIdx.x * blockDim.x + threadIdx.x;
    if (e < E) norm[e] = dinv[src[e]] * w[e] * dinv[dst[e]];
}

// C[M,N] = A[M,K] @ W[K,N], fp32 WMMA 16x16x4.
// One wave per 16x16 tile. Requires M%16==0, N%16==0, K%4==0 (true here:
// M=50000=3125*16, K in {128,64}, N in {64,32}) -> EXEC uniform per wave,
// no lane-level guards around the WMMA (ISA: EXEC must be all-ones).
__global__ __launch_bounds__(TPB)
void k_gemm_f32_wmma(const float* __restrict__ A, const float* __restrict__ W,
                     float* __restrict__ C, int M, int K, int N) {
    const int lane = threadIdx.x & 31;
    const int wave = threadIdx.x >> 5;
    const int wavesPerBlock = blockDim.x >> 5;
    const int colTiles = N >> 4;
    const int totalTiles = (M >> 4) * colTiles;

    int tile = blockIdx.x * wavesPerBlock + wave;
    if (tile >= totalTiles) return;            // whole-wave uniform exit

    const int rowTile = tile / colTiles;
    const int colTile = tile - rowTile * colTiles;
    const int half = lane >> 4;                // 0: lanes 0-15, 1: lanes 16-31
    const int lid  = lane & 15;
    const int aRow = rowTile * 16 + lid;       // A row held by this lane
    const int col  = colTile * 16 + lid;       // B/C/D column held by this lane

    v8f acc = {};
    for (int k0 = 0; k0 < K; k0 += 4) {
        const int ka = k0 + 2 * half;          // this half-wave's K pair
        v2f a, b;
        // A 16x4 layout: v0 = K=ka, v1 = K=ka+1 (contiguous in row-major A)
        a.x = A[(size_t)aRow * K + ka];
        a.y = A[(size_t)aRow * K + ka + 1];
        // B 4x16 layout: rows striped across lanes, v0 = row ka, v1 = row ka+1
        b.x = W[(size_t)ka * N + col];
        b.y = W[(size_t)(ka + 1) * N + col];
        acc = __builtin_amdgcn_wmma_f32_16x16x4_f32(
            /*neg_a=*/false, a, /*neg_b=*/false, b,
            /*c_mod=*/(short)0, acc, /*reuse_a=*/false, /*reuse_b=*/false);
    }

    // C/D layout: VGPR r -> M = r + 8*half, N = lid
    const int m0 = rowTile * 16 + 8 * half;
#pragma unroll
    for (int r = 0; r < 8; ++r)
        C[(size_t)(m0 + r) * N + col] = acc[r];
}

// Out[i,f] = snorm[i] * H[i,f]   (self-loop contribution; also initializes Out)
__global__ void k_agg_init(const float* __restrict__ H, const float* __restrict__ snorm,
                           float* __restrict__ Out, int N, int F) {
    long long t = (long long)blockIdx.x * blockDim.x + threadIdx.x;
    if (t < (long long)N * F) {
        int i = (int)(t / F);
        Out[t] = snorm[i] * H[t];
    }
}

// Out[dst[e],f] += norm[e] * H[src[e],f]
__global__ void k_scatter(const float* __restrict__ H, const float* __restrict__ norm,
                          const int* __restrict__ src, const int* __restrict__ dst,
                          float* __restrict__ Out, int E, int F) {
    long long t = (long long)blockIdx.x * blockDim.x + threadIdx.x;
    if (t >= (long long)E * F) return;
    int e = (int)(t / F);
    int f = (int)(t - (long long)e * F);
    float v = norm[e] * H[(size_t)src[e] * F + f];
    atomicAdd(&Out[(size_t)dst[e] * F + f], v);
}

// Out[i,f] = (Out[i,f] + b[f]) with optional ReLU
__global__ void k_finalize(float* __restrict__ Out, const float* __restrict__ b,
                           int N, int F, int relu) {
    long long t = (long long)blockIdx.x * blockDim.x + threadIdx.x;
    if (t < (long long)N * F) {
        int f = (int)(t % F);
        float v = Out[t] + b[f];
        Out[t] = relu ? fmaxf(v, 0.0f) : v;
    }
}

// outMean[f] = mean_i nodeEmb[i,f]; one block per feature column
__global__ void k_mean(const float* __restrict__ nodeEmb, float* __restrict__ outMean,
                       int N, int F) {
    __shared__ float sdata[TPB];
    const int f = blockIdx.x;
    float s = 0.0f;
    for (int i = threadIdx.x; i < N; i += blockDim.x)
        s += nodeEmb[(size_t)i * F + f];
    sdata[threadIdx.x] = s;
    __syncthreads();
    for (int stride = blockDim.x >> 1; stride > 0; stride >>= 1) {
        if (threadIdx.x < stride) sdata[threadIdx.x] += sdata[threadIdx.x + stride];
        __syncthreads();
    }
    if (threadIdx.x == 0) outMean[f] = sdata[0] / (float)N;
}

static inline int cdiv_ll(long long a, int b) { return (int)((a + b - 1) / b); }

extern "C" void kernel_launch(void* const* d_in, const int* in_sizes, int n_in,
                              void* d_out, int out_size, void* d_ws, size_t ws_size,
                              hipStream_t stream) {
    const float* x  = (const float*)d_in[0];
    const int*   ei = (const int*)  d_in[1];
    const float* ea = (const float*)d_in[2];
    const float* W1 = (const float*)d_in[3];
    const float* b1 = (const float*)d_in[4];
    const float* W2 = (const float*)d_in[5];
    const float* b2 = (const float*)d_in[6];
    const float* W3 = (const float*)d_in[7];
    const float* b3 = (const float*)d_in[8];

    const int E = in_sizes[2];        // edge_attr has E elements
    const int N = in_sizes[0] / 128;  // x is [N,128]
    const int* src = ei;              // edge_index[0,:]
    const int* dst = ei + E;          // edge_index[1,:]

    // Workspace layout (floats): deg[N] dinv[N] snorm[N] norm[E]
    //                            bufA[N*64] bufB[N*64] bufC[N*32]   (~36 MB)
    float* ws    = (float*)d_ws;
    float* deg   = ws;
    float* dinv  = deg   + N;
    float* snorm = dinv  + N;
    float* norm  = snorm + N;
    float* bufA  = norm  + E;
    float* bufB  = bufA  + (size_t)N * 64;
    float* bufC  = bufB  + (size_t)N * 64;

    float* outN    = (float*)d_out;           // [N,32] node embeddings
    float* outMean = outN + (size_t)N * 32;   // [1,32] graph embedding

    // ---- normalization -----------------------------------------------------
    k_init_deg<<<cdiv_ll(N, TPB), TPB, 0, stream>>>(deg, N);
    k_edge_deg<<<cdiv_ll(E, TPB), TPB, 0, stream>>>(dst, ea, deg, E);
    k_dinv    <<<cdiv_ll(N, TPB), TPB, 0, stream>>>(deg, dinv, snorm, N);
    k_norm    <<<cdiv_ll(E, TPB), TPB, 0, stream>>>(src, dst, ea, dinv, norm, E);

    const int wavesPerBlock = TPB / 32;
    auto gemm_grid = [&](int M, int Nc) {
        int tiles = (M / 16) * (Nc / 16);
        return cdiv_ll(tiles, wavesPerBlock);
    };

    // ---- layer 1: h1 = relu(agg(x @ W1) + b1) ------------------------------
    k_gemm_f32_wmma<<<gemm_grid(N, 64), TPB, 0, stream>>>(x, W1, bufA, N, 128, 64);
    k_agg_init<<<cdiv_ll((long long)N * 64, TPB), TPB, 0, stream>>>(bufA, snorm, bufB, N, 64);
    k_scatter <<<cdiv_ll((long long)E * 64, TPB), TPB, 0, stream>>>(bufA, norm, src, dst, bufB, E, 64);
    k_finalize<<<cdiv_ll((long long)N * 64, TPB), TPB, 0, stream>>>(bufB, b1, N, 64, 1);

    // ---- layer 2: h2 = relu(agg(h1 @ W2) + b2) -----------------------------
    k_gemm_f32_wmma<<<gemm_grid(N, 64), TPB, 0, stream>>>(bufB, W2, bufA, N, 64, 64);
    k_agg_init<<<cdiv_ll((long long)N * 64, TPB), TPB, 0, stream>>>(bufA, snorm, bufB, N, 64);
    k_scatter <<<cdiv_ll((long long)E * 64, TPB), TPB, 0, stream>>>(bufA, norm, src, dst, bufB, E, 64);
    k_finalize<<<cdiv_ll((long long)N * 64, TPB), TPB, 0, stream>>>(bufB, b2, N, 64, 1);

    // ---- layer 3: out = agg(h2 @ W3) + b3 (no relu) ------------------------
    k_gemm_f32_wmma<<<gemm_grid(N, 32), TPB, 0, stream>>>(bufB, W3, bufC, N, 64, 32);
    k_agg_init<<<cdiv_ll((long long)N * 32, TPB), TPB, 0, stream>>>(bufC, snorm, outN, N, 32);
    k_scatter <<<cdiv_ll((long long)E * 32, TPB), TPB, 0, stream>>>(bufC, norm, src, dst, outN, E, 32);
    k_finalize<<<cdiv_ll((long long)N * 32, TPB), TPB, 0, stream>>>(outN, b3, N, 32, 0);

    // ---- global mean pool --------------------------------------------------
    k_mean<<<32, TPB, 0, stream>>>(outN, outMean, N, 32);
}